// SAGE_62388694942260
// MI455X (gfx1250) — compile-verified
//
#include <hip/hip_runtime.h>

#define N_NODES  50000
#define N_EDGES  800000
#define HID      128
#define OUT_SIZE 47
#define N_LAYERS 3
#define MAXK     32
#define APAD     4      // LDS row padding (floats): keeps float4 align, kills bank conflicts

typedef float v2f __attribute__((ext_vector_type(2)));
typedef float v8f __attribute__((ext_vector_type(8)));

__device__ __forceinline__ unsigned umax_u(unsigned a, unsigned b) { return a > b ? a : b; }

// ---------------------------------------------------------------------------
// Fused GEMM:  OUT[M x Nout] = A[M x 128] @ W[Nout x 128]^T
//                            (+ A2 @ W2^T when DUAL) + bias
// Block = 256 threads = 8 waves; block stages its 16-row M-tile(s) in LDS
// (float4, coalesced); wave w computes N-tile [16w,16w+16) with
// V_WMMA_F32_16X16X4_F32 (fp32-exact, matches reference precision).
// NGUARD=false: Nout==128, unconditional loads/stores (no EXEC churn).
// NGUARD=true : Nout==47, clamped addresses + cndmask zeroing (branchless).
// ---------------------------------------------------------------------------
template <bool DUAL, bool NGUARD>
__global__ __launch_bounds__(256) void wmma_gemm_kernel(
    const float* __restrict__ A, const float* __restrict__ W,
    const float* __restrict__ A2, const float* __restrict__ W2,
    const float* __restrict__ bias, float* __restrict__ OUT, int Nout) {
  __shared__ float sA[16][HID + APAD];
  __shared__ float sA2[DUAL ? 16 : 1][DUAL ? (HID + APAD) : 1];

  const int tm  = blockIdx.x * 16;
  const int tid = threadIdx.x;

  // Stage 16 x 128 A-tile(s): 512 float4s, 2 per thread, fully coalesced.
  for (int i = tid; i < 16 * (HID / 4); i += 256) {
    const int r = i >> 5;            // 32 float4 per row
    const int c4 = (i & 31) * 4;
    *(float4*)&sA[r][c4] = *(const float4*)(A + (size_t)(tm + r) * HID + c4);
    if (DUAL)
      *(float4*)&sA2[r][c4] = *(const float4*)(A2 + (size_t)(tm + r) * HID + c4);
  }
  __syncthreads();

  const int wave = tid >> 5;
  const int lane = tid & 31;
  const int half = lane >> 4;        // K-pair select: lanes 0-15 -> K{0,1}, 16-31 -> K{2,3}
  const int q    = lane & 15;        // M (A-frag) / N (B-frag, C)
  const int n    = wave * 16 + q;
  const bool nok = NGUARD ? (n < Nout) : true;
  const int nc   = NGUARD ? (nok ? n : Nout - 1) : n;   // clamped, always in-bounds

  const float* wp  = W + (size_t)nc * HID;
  const float* wp2 = DUAL ? (W2 + (size_t)nc * HID) : nullptr;

  v8f acc = {};
#pragma unroll 4
  for (int k0 = 0; k0 < HID; k0 += 4) {
    const int kb = k0 + half * 2;
    const float2 af = *(const float2*)&sA[q][kb];       // ds_load_b64
    float2 bf = *(const float2*)(wp + kb);              // global_load_b64
    if (NGUARD && !nok) { bf.x = 0.0f; bf.y = 0.0f; }   // v_cndmask, no EXEC churn
    v2f a, b;
    a[0] = af.x; a[1] = af.y;
    b[0] = bf.x; b[1] = bf.y;
    acc = __builtin_amdgcn_wmma_f32_16x16x4_f32(false, a, false, b,
                                                (short)0, acc, false, false);
    if (DUAL) {
      const float2 af2 = *(const float2*)&sA2[q][kb];
      const float2 bf2 = *(const float2*)(wp2 + kb);
      v2f a2, b2;
      a2[0] = af2.x; a2[1] = af2.y;
      b2[0] = bf2.x; b2[1] = bf2.y;
      acc = __builtin_amdgcn_wmma_f32_16x16x4_f32(false, a2, false, b2,
                                                  (short)0, acc, false, false);
    }
  }

  if (nok) {
    const float bv = bias[n];
    const int mbase = tm + half * 8;  // C VGPR r: lanes 0-15 -> M=r, lanes 16-31 -> M=r+8
#pragma unroll
    for (int r = 0; r < 8; ++r)
      OUT[(size_t)(mbase + r) * Nout + n] = acc[r] + bv;
  }
}

// ---------------------------------------------------------------------------
// MaxK: keep top-32 of 128 per row, zero the rest. One wave per row; lane
// holds cols {lane, lane+32, lane+64, lane+96}. 32 rounds of wave-max
// extraction over order-preserving uint keys finds the 32nd-largest value.
// ---------------------------------------------------------------------------
__global__ __launch_bounds__(256) void maxk_kernel(const float* __restrict__ in,
                                                   float* __restrict__ out,
                                                   int nrows) {
  const int wave = threadIdx.x >> 5;
  const int lane = threadIdx.x & 31;
  const int row  = blockIdx.x * 8 + wave;
  if (row >= nrows) return;

  const float* rp = in + (size_t)row * HID;
  float v[4];
  unsigned key[4], rem[4];
#pragma unroll
  for (int j = 0; j < 4; ++j) {
    const float x = rp[lane + 32 * j];
    v[j] = x;
    const unsigned u = __float_as_uint(x);
    key[j] = (u & 0x80000000u) ? ~u : (u | 0x80000000u);
    rem[j] = key[j];
  }

  unsigned thresh = 0;
  for (int it = 0; it < MAXK; ++it) {
    unsigned lm = umax_u(umax_u(rem[0], rem[1]), umax_u(rem[2], rem[3]));
    unsigned wm = lm;
#pragma unroll
    for (int off = 16; off > 0; off >>= 1)
      wm = umax_u(wm, (unsigned)__shfl_xor((int)wm, off, 32));
    const unsigned long long bal = __ballot(lm == wm);
    const int first = __ffsll((unsigned long long)bal) - 1;
    if (lane == first) {             // remove exactly one instance of current max
      if (rem[0] == wm)      rem[0] = 0;
      else if (rem[1] == wm) rem[1] = 0;
      else if (rem[2] == wm) rem[2] = 0;
      else                   rem[3] = 0;
    }
    thresh = wm;
  }

  float* op = out + (size_t)row * HID;
#pragma unroll
  for (int j = 0; j < 4; ++j)
    op[lane + 32 * j] = (key[j] >= thresh) ? v[j] : 0.0f;
}

// ---------------------------------------------------------------------------
// Edge scatter: each wave handles 4 consecutive edges; lane carries a float4
// slice of the 512B row. Source rows are prefetched up front
// (global_prefetch_b8) to hide the irregular-gather latency; then float4
// gather + 4x atomicAdd scatter per edge; lane 0 bumps the degree counter.
// 800000 edges / (8 waves * 4 edges) = 25000 blocks exactly.
// ---------------------------------------------------------------------------
#define EDGES_PER_WAVE 4
__global__ __launch_bounds__(256) void scatter_kernel(
    const float* __restrict__ h, const int* __restrict__ src,
    const int* __restrict__ dst, float* __restrict__ agg,
    float* __restrict__ deg, int nedges) {
  const int wave = threadIdx.x >> 5;
  const int lane = threadIdx.x & 31;
  const int base = (blockIdx.x * 8 + wave) * EDGES_PER_WAVE;

  int s[EDGES_PER_WAVE], d[EDGES_PER_WAVE];
#pragma unroll
  for (int t = 0; t < EDGES_PER_WAVE; ++t) {
    s[t] = src[base + t];
    d[t] = dst[base + t];
  }
#pragma unroll
  for (int t = 0; t < EDGES_PER_WAVE; ++t)
    __builtin_prefetch(h + (size_t)s[t] * HID + lane * 4, 0, 3);

#pragma unroll
  for (int t = 0; t < EDGES_PER_WAVE; ++t) {
    const float4 m = *(const float4*)(h + (size_t)s[t] * HID + lane * 4);
    float* ap = agg + (size_t)d[t] * HID + lane * 4;
    atomicAdd(ap + 0, m.x);
    atomicAdd(ap + 1, m.y);
    atomicAdd(ap + 2, m.z);
    atomicAdd(ap + 3, m.w);
    if (lane == 0) atomicAdd(deg + d[t], 1.0f);
  }
}

__global__ __launch_bounds__(256) void zero_kernel(float* __restrict__ p, int n) {
  const int i = blockIdx.x * 256 + threadIdx.x;
  if (i < n) p[i] = 0.0f;
}

// agg[i][:] /= max(deg[i], 1)
__global__ __launch_bounds__(256) void norm_kernel(float* __restrict__ agg,
                                                   const float* __restrict__ deg,
                                                   int nnodes) {
  const int i = blockIdx.x * 256 + threadIdx.x;
  if (i >= nnodes * 32) return;
  const int node = i >> 5;
  const int c4 = (i & 31) * 4;
  const float r = 1.0f / fmaxf(deg[node], 1.0f);
  float4* p = (float4*)(agg + (size_t)node * HID + c4);
  float4 x = *p;
  x.x *= r; x.y *= r; x.z *= r; x.w *= r;
  *p = x;
}

extern "C" void kernel_launch(void* const* d_in, const int* in_sizes, int n_in,
                              void* d_out, int out_size, void* d_ws, size_t ws_size,
                              hipStream_t stream) {
  const float* x        = (const float*)d_in[0];
  const int*   src      = (const int*)d_in[1];
  const int*   dst      = (const int*)d_in[2];
  const float* lin_in_w = (const float*)d_in[3];
  const float* lin_in_b = (const float*)d_in[4];
  const float* self_w   = (const float*)d_in[5];
  const float* neigh_w  = (const float*)d_in[6];
  const float* conv_b   = (const float*)d_in[7];
  const float* lout_w   = (const float*)d_in[8];
  const float* lout_b   = (const float*)d_in[9];
  float* out = (float*)d_out;

  float* hA  = (float*)d_ws;                      // [N, HID] layer activations
  float* hB  = hA  + (size_t)N_NODES * HID;       // [N, HID] post-MaxK
  float* agg = hB  + (size_t)N_NODES * HID;       // [N, HID] neighbor sum/mean
  float* deg = agg + (size_t)N_NODES * HID;       // [N]

  const int mtiles = N_NODES / 16;                // 3125 exactly

  // h = x @ lin_in_w^T + b
  wmma_gemm_kernel<false, false><<<dim3(mtiles), 256, 0, stream>>>(
      x, lin_in_w, nullptr, nullptr, lin_in_b, hA, HID);

  for (int l = 0; l < N_LAYERS; ++l) {
    maxk_kernel<<<N_NODES / 8, 256, 0, stream>>>(hA, hB, N_NODES);
    zero_kernel<<<(N_NODES * HID + N_NODES + 255) / 256, 256, 0, stream>>>(
        agg, N_NODES * HID + N_NODES);            // agg and deg are contiguous
    scatter_kernel<<<N_EDGES / (8 * EDGES_PER_WAVE), 256, 0, stream>>>(
        hB, src, dst, agg, deg, N_EDGES);
    norm_kernel<<<(N_NODES * 32 + 255) / 256, 256, 0, stream>>>(agg, deg, N_NODES);
    // h = maxk(h) @ self_w^T + mean_agg @ neigh_w^T + b   (fused dual GEMM)
    wmma_gemm_kernel<true, false><<<dim3(mtiles), 256, 0, stream>>>(
        hB, self_w + (size_t)l * HID * HID, agg, neigh_w + (size_t)l * HID * HID,
        conv_b + (size_t)l * HID, hA, HID);
  }

  // out = h @ lin_out_w^T + b   (Nout = 47, clamped/branchless guard)
  wmma_gemm_kernel<false, true><<<dim3(mtiles), 256, 0, stream>>>(
      hA, lout_w, nullptr, nullptr, lout_b, out, OUT_SIZE);
}